// Castro2025Model_79259326481053
// MI455X (gfx1250) — compile-verified
//
#include <hip/hip_runtime.h>
#include <hip/hip_bf16.h>
#include <stdint.h>

#define B_SESS 8192
#define T_TR   512
#define NA     4
#define SPB    32                      // sessions per block
#define TC     16                      // timesteps per staged chunk
#define NCHUNK (T_TR / TC)             // 32
#define REC_B  32                      // bytes per (session, step) record: 8 f32
#define SESS_LDS_STRIDE 544            // 512B payload + 32B pad -> conflict-free banks
#define BUF_BYTES (SPB * SESS_LDS_STRIDE)
#define THREADS (SPB * NA)             // 128 = 4 wave32

// ---- gfx1250 async-tensor data movement (ASYNCcnt-tracked DMA to LDS) ----
__device__ __forceinline__ void async_load16(unsigned ldsOff, unsigned long long gaddr) {
  asm volatile("global_load_async_to_lds_b128 %0, %1, off"
               :: "v"(ldsOff), "v"(gaddr) : "memory");
}
__device__ __forceinline__ void wait_async0() {
  asm volatile("s_wait_asynccnt 0" ::: "memory");
}

// ---- quad (4-lane = 1 session) exchanges via DPP quad_perm (VALU-speed) ----
// 0xB1 = quad_perm[1,0,3,2] (xor 1) ; 0x4E = quad_perm[2,3,0,1] (xor 2)
template <int CTRL>
__device__ __forceinline__ float dpp_qpermf(float v) {
  return __int_as_float(
      __builtin_amdgcn_update_dpp(0, __float_as_int(v), CTRL, 0xf, 0xf, true));
}
template <int CTRL>
__device__ __forceinline__ int dpp_qpermi(int v) {
  return __builtin_amdgcn_update_dpp(0, v, CTRL, 0xf, 0xf, true);
}
__device__ __forceinline__ float quad_sum(float v) {
  v += dpp_qpermf<0xB1>(v);
  v += dpp_qpermf<0x4E>(v);
  return v;
}
__device__ __forceinline__ float quad_max(float v) {
  v = fmaxf(v, dpp_qpermf<0xB1>(v));
  v = fmaxf(v, dpp_qpermf<0x4E>(v));
  return v;
}
__device__ __forceinline__ int quad_isum(int v) {
  v += dpp_qpermi<0xB1>(v);
  v += dpp_qpermi<0x4E>(v);
  return v;
}

__device__ __forceinline__ float softplusf_(float x) {
  return fmaxf(x, 0.0f) + log1pf(expf(-fabsf(x)));
}
__device__ __forceinline__ float sigmoidf_(float x) {
  return 1.0f / (1.0f + expf(-x));
}
__device__ __forceinline__ float clampf_(float x, float lo, float hi) {
  return fminf(fmaxf(x, lo), hi);
}

struct __align__(16) F4 { float x, y, z, w; };

__global__ __launch_bounds__(THREADS)
void castro_scan_kernel(const float* __restrict__ inputs,
                        const float* __restrict__ praw,
                        float* __restrict__ out) {
  __shared__ char smem[2 * BUF_BYTES];

  const int tid  = threadIdx.x;
  const int s0   = blockIdx.x * SPB;
  const int sRel = tid >> 2;     // local session 0..31
  const int aIdx = tid & 3;      // my action lane 0..3
  const int sess = s0 + sRel;

  // ---- parameter transform (cheap, per-thread) ----
  float p[13];
  #pragma unroll
  for (int i = 0; i < 13; ++i) p[i] = praw[i];
  const float beta_r = clampf_(softplusf_(p[0]), 0.01f, 20.0f);
  const float lapse  = clampf_(sigmoidf_(p[1]), 0.01f, 0.99f);
  const float prior  = clampf_(softplusf_(p[2]), 0.01f, 0.99f);
  const float alpha  = clampf_(sigmoidf_(p[3]), 0.01f, 0.99f);
  const float decay  = clampf_(sigmoidf_(p[4]), 0.01f, 0.99f);
  const float ab1 = p[5], ab2 = p[6];
  const float pers = softplusf_(p[7]);
  const float sw   = p[8];
  const float gamma_ = softplusf_(p[10]);
  const float temp   = clampf_(softplusf_(p[11]) + 1e-6f, 1e-6f, 100.0f);
  const float beta_p = softplusf_(p[12]);
  const float qscale = beta_r / temp;

  const unsigned long long inBase =
      (unsigned long long)(uintptr_t)inputs +
      (unsigned long long)s0 * (unsigned long long)(T_TR * REC_B);
  // low 32 bits of a generic pointer to LDS == LDS byte offset (aperture encoding)
  const unsigned ldsBase = (unsigned)(uintptr_t)(&smem[0]);

  // ---- stage chunk 0 into buffer 0 (coalesced: 32 lanes cover one 512B session row) ----
  {
    const unsigned seg = (unsigned)(tid & 31);
    const unsigned sB  = (unsigned)(tid >> 5);
    #pragma unroll
    for (int i = 0; i < 8; ++i) {
      const unsigned sr = sB + (unsigned)i * 4u;
      const unsigned long long g =
          inBase + (unsigned long long)sr * (T_TR * REC_B) + seg * 16u;
      const unsigned l = ldsBase + sr * SESS_LDS_STRIDE + seg * 16u;
      async_load16(l, g);
    }
  }
  wait_async0();
  __syncthreads();

  // ---- per-lane state ----
  float q    = prior;   // my action's Q
  float cum  = 0.0f;    // my action's choice count
  float tsls = 0.0f;    // trials since last switch (replicated per session)
  float expl = alpha;   // exploration mix (replicated)
  int   oldc = -1;      // previous choice (replicated)

  float* outp = out + (size_t)sess * (T_TR * NA);

  for (int c = 0; c < NCHUNK; ++c) {
    const int buf = c & 1;

    // prefetch next chunk into the other buffer (overlaps with compute below)
    if (c + 1 < NCHUNK) {
      const unsigned seg = (unsigned)(tid & 31);
      const unsigned sB  = (unsigned)(tid >> 5);
      const unsigned t0b = (unsigned)(c + 1) * (TC * REC_B);
      #pragma unroll
      for (int i = 0; i < 8; ++i) {
        const unsigned sr = sB + (unsigned)i * 4u;
        const unsigned long long g =
            inBase + (unsigned long long)sr * (T_TR * REC_B) + t0b + seg * 16u;
        const unsigned l =
            ldsBase + (unsigned)(buf ^ 1) * BUF_BYTES + sr * SESS_LDS_STRIDE + seg * 16u;
        async_load16(l, g);
      }
    }

    const char* mybase = smem + buf * BUF_BYTES + sRel * SESS_LDS_STRIDE;

    // 4 groups of 4 steps; each group ends with a quad 4x4 transpose + b128 store
    for (int g = 0; g < TC / 4; ++g) {
      float o[4];
      #pragma unroll
      for (int j = 0; j < 4; ++j) {
        const char* rec = mybase + (g * 4 + j) * REC_B;
        const float myA = *(const float*)(rec + (aIdx << 2)); // one-hot action, my lane
        const float rv  = *(const float*)(rec + 16);          // reward (broadcast elem 4)
        const bool  mine = myA > 0.5f;
        const float ohf  = mine ? 1.0f : 0.0f;
        const int   cc   = quad_isum(mine ? aIdx : 0);        // argmax of one-hot

        // Q overwrite-error update; q[cc] is the chosen lane's own q,
        // so delta needs no cross-lane reduction.
        const float target = rv - gamma_ * (1.0f - rv);
        q = mine ? (q + (target - q)) : q;

        const bool same = (cc == oldc);
        tsls = same ? (tsls + 1.0f) : 0.0f;
        expl *= (1.0f - 1e-3f);
        cum  += ohf;

        // mean-mixing exploration + decay
        const float qm = 0.25f * quad_sum(q);
        q = (1.0f - expl) * q + expl * qm;
        q *= decay;

        // lapse-mixed softmax with count bonus
        const float lg  = qscale * q + beta_p * log1pf(cum);
        const float mx  = quad_max(lg);
        const float e   = expf(lg - mx);
        const float den = quad_sum(e);
        const float prob = (1.0f - lapse) * (e / den) + 0.25f * lapse;
        float val = logf(prob);

        // bonuses
        if (mine) val += (same ? pers : sw) + log1pf(tsls);
        if (oldc >= 0 && aIdx == oldc) val += ab1;
        if (aIdx == ((cc + 2) & 3))    val += ab2;

        o[j] = val;
        oldc = cc;
      }

      // quad 4x4 transpose: lane a ends with step (t0+a)'s 4 action logits
      {
        const bool b0 = (aIdx & 1) != 0;
        float x01 = b0 ? o[0] : o[1];
        float x23 = b0 ? o[2] : o[3];
        x01 = dpp_qpermf<0xB1>(x01);
        x23 = dpp_qpermf<0xB1>(x23);
        o[0] = b0 ? x01 : o[0];  o[1] = b0 ? o[1] : x01;
        o[2] = b0 ? x23 : o[2];  o[3] = b0 ? o[3] : x23;
      }
      {
        const bool b1 = (aIdx & 2) != 0;
        float x02 = b1 ? o[0] : o[2];
        float x13 = b1 ? o[1] : o[3];
        x02 = dpp_qpermf<0x4E>(x02);
        x13 = dpp_qpermf<0x4E>(x13);
        o[0] = b1 ? x02 : o[0];  o[2] = b1 ? o[2] : x02;
        o[1] = b1 ? x13 : o[1];  o[3] = b1 ? o[3] : x13;
      }

      const int tstep = c * TC + g * 4 + aIdx;   // step this lane now owns
      F4 v4; v4.x = o[0]; v4.y = o[1]; v4.z = o[2]; v4.w = o[3];
      *(F4*)(outp + (tstep << 2)) = v4;          // 16B store, 64B runs per session
    }

    wait_async0();     // my wave's prefetch landed
    __syncthreads();   // everyone's landed; buffers safe to swap
  }
}

extern "C" void kernel_launch(void* const* d_in, const int* in_sizes, int n_in,
                              void* d_out, int out_size, void* d_ws, size_t ws_size,
                              hipStream_t stream) {
  const float* inputs = (const float*)d_in[0];   // [8192, 512, 8] f32
  const float* praw   = (const float*)d_in[1];   // [13] f32
  float* out          = (float*)d_out;           // [8192, 512, 4] f32
  (void)in_sizes; (void)n_in; (void)out_size; (void)d_ws; (void)ws_size;

  dim3 grid(B_SESS / SPB);   // 256 blocks
  dim3 block(THREADS);       // 128 threads = 4 wave32, 32 sessions x 4 actions
  hipLaunchKernelGGL(castro_scan_kernel, grid, block, 0, stream, inputs, praw, out);
}